// projector_48627619725569
// MI455X (gfx1250) — compile-verified
//
#include <hip/hip_runtime.h>

// Fan-beam CT forward projection for gfx1250 (MI455X).
// Gather-bound workload; image (8 MB) is L2-resident (192 MB L2).
// Per-wave: 16 detectors -> rows of a 16x16x4 f32 WMMA; A = interpolated
// values (16 det x 4 samples), B = all-ones 4x16, C accumulates the
// per-detector running sum on the matrix pipe while the VALU does the
// bilinear math. Neighbor pairs adjacent in x are fetched as single b64
// gathers; image-border validity is folded into the weights via selects
// computed once per sample and shared across all 8 batch images.

typedef __attribute__((ext_vector_type(2))) float v2f;
typedef v2f __attribute__((aligned(4))) v2f_a4;   // allow dword-aligned b64 loads
typedef __attribute__((ext_vector_type(8))) float v8f;

#define IMG_N        512
#define NUM_DET      768
#define NUM_SAMPLES  768
#define NUM_VIEWS    360
#define NBATCH       8

__global__ __launch_bounds__(256) void fanbeam_fp_wmma_kernel(
    const float* __restrict__ image,   // [8, 512, 512]
    const float* __restrict__ views,   // [360]
    float* __restrict__ out)           // [8, 360, 768]
{
    constexpr float PIX     = 0.7433f;
    constexpr float DETSZ   = 1.2858f;
    constexpr float ISO     = 595.0f;
    constexpr float SDD     = 595.0f + 490.6f;
    constexpr float INV_PIX = 1.0f / PIX;
    constexpr float HALF_N  = (IMG_N - 1) * 0.5f;
    constexpr float FOV     = IMG_N * PIX * 0.70710678f;
    constexpr float T0      = ISO - FOV;
    constexpr float DT      = 2.0f * FOV / NUM_SAMPLES;
    constexpr float DGAMMA  = DETSZ / SDD;
    constexpr float PI_F    = 3.14159265358979323846f;

    const int view    = blockIdx.x;
    const int wave    = threadIdx.x >> 5;
    const int lane    = threadIdx.x & 31;
    const int m       = lane & 15;      // detector row within the 16-det tile
    const int kh      = lane >> 4;      // K half: lanes 0-15 -> K=0,1 ; 16-31 -> K=2,3
    const int detBase = (blockIdx.y * 8 + wave) * 16;
    const int det     = detBase + m;

    // Per-ray geometry (computed once per thread).
    const float beta  = views[view];
    const float gamma = ((float)det - (NUM_DET - 1) * 0.5f) * DGAMMA;
    const float sx    = ISO * cosf(beta);
    const float sy    = ISO * sinf(beta);
    const float ang   = beta + PI_F + gamma;
    const float dirx  = cosf(ang);
    const float diry  = sinf(ang);

    // Per-batch WMMA accumulators: 8 x v8f = 64 VGPRs.
    v8f acc[NBATCH];
#pragma unroll
    for (int b = 0; b < NBATCH; ++b) {
        v8f z = {};
        acc[b] = z;
    }

    // All-ones B matrix (4x16): every element 1.0 -> both B VGPRs = 1.0.
    v2f bones = {1.0f, 1.0f};

    for (int s0 = 0; s0 < NUM_SAMPLES; s0 += 4) {
        const int sA = s0 + (kh << 1);   // this lane's first sample in the chunk

        // Per-sample: two row base offsets + four pair-weights; shared
        // across all 8 batches.
        float wgt[2][4];
        int   off0[2];    // row y0 (clamped) base offset of the x-pair
        int   off1[2];    // row y1 (clamped) base offset of the x-pair
#pragma unroll
        for (int q = 0; q < 2; ++q) {
            const float t  = T0 + ((float)(sA + q) + 0.5f) * DT;
            const float fx = (sx + dirx * t) * INV_PIX + HALF_N;
            const float fy = (sy + diry * t) * INV_PIX + HALF_N;
            const float x0f = floorf(fx);
            const float y0f = floorf(fy);
            const float wx  = fx - x0f;
            const float wy  = fy - y0f;
            const int x0 = (int)x0f;
            const int y0 = (int)y0f;
            const int y1 = y0 + 1;

            // x direction: single b64 gather at xm=clamp(x0,0,N-2); border
            // validity folded into the pair weights (wx_lo -> loaded .x,
            // wx_hi -> loaded .y).
            const int  xm = min(max(x0, 0), IMG_N - 2);
            const bool xin = (x0 == xm);          // 0 <= x0 <= N-2: both valid
            const bool xlo = (x0 == -1);          // only x1=0 valid -> .x
            const bool xhi = (x0 == IMG_N - 1);   // only x0=N-1 valid -> .y
            const float wx_lo = xin ? (1.0f - wx) : (xlo ? wx : 0.0f);
            const float wx_hi = xin ? wx : (xhi ? (1.0f - wx) : 0.0f);

            // y direction: per-row validity into the weights, clamped rows.
            const bool vy0 = (unsigned)y0 < (unsigned)IMG_N;
            const bool vy1 = (unsigned)y1 < (unsigned)IMG_N;
            const float wy0 = vy0 ? (1.0f - wy) : 0.0f;
            const float wy1 = vy1 ? wy : 0.0f;
            const int y0c = min(max(y0, 0), IMG_N - 1);
            const int y1c = min(max(y1, 0), IMG_N - 1);

            off0[q] = y0c * IMG_N + xm;
            off1[q] = y1c * IMG_N + xm;
            wgt[q][0] = wx_lo * wy0;   // row0 .x
            wgt[q][1] = wx_hi * wy0;   // row0 .y
            wgt[q][2] = wx_lo * wy1;   // row1 .x
            wgt[q][3] = wx_hi * wy1;   // row1 .y
        }

        // Gather (2 x b64 per sample) + interpolate per batch, then reduce
        // across samples via WMMA: D[m][n] = sum_k A[m][k] + C[m][n].
#pragma unroll
        for (int b = 0; b < NBATCH; ++b) {
            const float* __restrict__ img = image + (size_t)b * (IMG_N * IMG_N);
            const v2f r0a = *(const v2f_a4*)(img + off0[0]);
            const v2f r1a = *(const v2f_a4*)(img + off1[0]);
            const v2f r0b = *(const v2f_a4*)(img + off0[1]);
            const v2f r1b = *(const v2f_a4*)(img + off1[1]);
            v2f a;
            a.x = r0a.x * wgt[0][0] + r0a.y * wgt[0][1] +
                  r1a.x * wgt[0][2] + r1a.y * wgt[0][3];
            a.y = r0b.x * wgt[1][0] + r0b.y * wgt[1][1] +
                  r1b.x * wgt[1][2] + r1b.y * wgt[1][3];
            acc[b] = __builtin_amdgcn_wmma_f32_16x16x4_f32(
                /*neg_a=*/false, a, /*neg_b=*/false, bones,
                /*c_mod=*/(short)0, acc[b], /*reuse_a=*/false, /*reuse_b=*/false);
        }
    }

    // D layout (16x16 f32 C/D): VGPR r, lane 0  -> D[r][0]   (det detBase+r)
    //                           VGPR r, lane 16 -> D[8+r][0] (det detBase+8+r)
    // All columns are identical (B = ones), so lanes 0 and 16 carry the sums.
    if ((lane & 15) == 0) {
        const int dbase = detBase + (kh << 3);
#pragma unroll
        for (int b = 0; b < NBATCH; ++b) {
#pragma unroll
            for (int r = 0; r < 8; ++r) {
                out[((size_t)b * NUM_VIEWS + view) * NUM_DET + (dbase + r)] =
                    acc[b][r] * DT;
            }
        }
    }
}

extern "C" void kernel_launch(void* const* d_in, const int* in_sizes, int n_in,
                              void* d_out, int out_size, void* d_ws, size_t ws_size,
                              hipStream_t stream) {
    const float* image = (const float*)d_in[0];   // [8, 512, 512] f32
    const float* views = (const float*)d_in[1];   // [360] f32
    float* out = (float*)d_out;                   // [8, 360, 768] f32

    // 360 views x 6 groups-of-128-detectors; 256 threads = 8 wave32s,
    // each wave owns a 16-detector WMMA tile.
    dim3 grid(NUM_VIEWS, NUM_DET / 128, 1);
    fanbeam_fp_wmma_kernel<<<grid, dim3(256, 1, 1), 0, stream>>>(image, views, out);
}